// SelfAttention_51084341018717
// MI455X (gfx1250) — compile-verified
//
#include <hip/hip_runtime.h>

typedef __attribute__((ext_vector_type(2))) float v2f;
typedef __attribute__((ext_vector_type(8))) float v8f;

#define WMMA_F32(a, b, c) \
  __builtin_amdgcn_wmma_f32_16x16x4_f32(false, (a), false, (b), (short)0, (c), false, false)

constexpr int Bn  = 24;    // batch
constexpr int DM  = 128;   // d_model
constexpr int Hn  = 8;     // heads
constexpr int DKn = 16;    // d_k
constexpr int Ln  = 512;   // sequence length
constexpr int LT  = Ln / 16;   // 32 tiles of 16 along L
constexpr int STR = 516;       // padded LDS row stride (floats): 516%64==4 -> conflict-free cols

// ---------------------------------------------------------------------------
// Kernel 1: one workgroup per (h,b). Computes Q,K,V into LDS, then per-column
// (query-axis) softmax'd attention, writing head[b][h*16+v][m] to workspace.
// ---------------------------------------------------------------------------
__global__ __launch_bounds__(256) void attn_hb_kernel(
    const float* __restrict__ x,    // [B][128][512]
    const float* __restrict__ Wq,   // [H][B][16][128]
    const float* __restrict__ Wk,
    const float* __restrict__ Wv,
    float* __restrict__ head)       // [B][128][512] (workspace)
{
  extern __shared__ float smem[];               // 3 * 16*516 floats = 99072 B
  float* sQ = smem;
  float* sK = smem + DKn * STR;
  float* sV = smem + 2 * DKn * STR;

  const int hb   = blockIdx.x;                  // 0..H*B-1
  const int h    = hb / Bn;
  const int b    = hb % Bn;
  const int tid  = threadIdx.x;
  const int wave = tid >> 5;                    // 8 waves
  const int lane = tid & 31;
  const int lh   = lane & 15;
  const int hi   = (lane >> 4) & 1;             // 0: lanes 0-15, 1: lanes 16-31

  const float* xb   = x + (size_t)b * DM * Ln;
  const size_t woff = ((size_t)h * Bn + b) * (size_t)(DKn * DM);

  // ---- Phase A: Q/K/V = W[16,128] @ x[128,512], f32 WMMA, K-chunks of 4 ----
  for (int job = wave; job < 3 * LT; job += 8) {
    const int p  = job / LT;                    // 0:Q 1:K 2:V
    const int lt = job - p * LT;
    const float* W = (p == 0 ? Wq : (p == 1 ? Wk : Wv)) + woff;
    float* S = (p == 0 ? sQ : (p == 1 ? sK : sV));
    const int l0 = lt * 16;
    v8f acc = {0.f, 0.f, 0.f, 0.f, 0.f, 0.f, 0.f, 0.f};
    for (int d0 = 0; d0 < DM; d0 += 4) {
      const int ad = d0 + 2 * hi;
      v2f a;  a.x  = W[lh * DM + ad];           a.y  = W[lh * DM + ad + 1];
      v2f bb; bb.x = xb[ad * Ln + l0 + lh];     bb.y = xb[(ad + 1) * Ln + l0 + lh];
      acc = WMMA_F32(a, bb, acc);
    }
#pragma unroll
    for (int r = 0; r < 8; ++r)                 // D: lane holds col l0+lh, rows r(+8)
      S[(r + 8 * hi) * STR + l0 + lh] = acc[r];
  }
  __syncthreads();

  float* outp = head + ((size_t)b * DM + (size_t)h * DKn) * Ln;

  // ---- Phase B: per m-tile: softmax over l (columns of S), heads = V @ P ----
  for (int mt = wave; mt < LT; mt += 8) {
    const int m0 = mt * 16;

    // Pass 1: online column max + exp-sum over all l
    float mrun = -1e30f, srun = 0.f;
    for (int lt2 = 0; lt2 < LT; ++lt2) {
      const int l0 = lt2 * 16;
      v8f s = {0.f, 0.f, 0.f, 0.f, 0.f, 0.f, 0.f, 0.f};
#pragma unroll
      for (int kc = 0; kc < 4; ++kc) {
        const int k0 = kc * 4 + 2 * hi;
        v2f a;  a.x  = sQ[k0 * STR + l0 + lh];  a.y  = sQ[(k0 + 1) * STR + l0 + lh];
        v2f bb; bb.x = sK[k0 * STR + m0 + lh];  bb.y = sK[(k0 + 1) * STR + m0 + lh];
        s = WMMA_F32(a, bb, s);                 // S[l,m] tile, lane = column m
      }
      float tmax = -1e30f;
#pragma unroll
      for (int r = 0; r < 8; ++r) { float v = s[r] * 0.25f; s[r] = v; tmax = fmaxf(tmax, v); }
      tmax = fmaxf(tmax, __shfl_xor(tmax, 16, 32));   // combine lane pair (rows 0-7 / 8-15)
      const float newm = fmaxf(mrun, tmax);
      float psum = 0.f;
#pragma unroll
      for (int r = 0; r < 8; ++r) psum += __expf(s[r] - newm);
      psum += __shfl_xor(psum, 16, 32);
      srun = srun * __expf(mrun - newm) + psum;
      mrun = newm;
    }
    const float inv = 1.0f / srun;

    // Pass 2: recompute tiles, P = exp(s - max), heads += V @ P
    v8f hacc = {0.f, 0.f, 0.f, 0.f, 0.f, 0.f, 0.f, 0.f};
    for (int lt2 = 0; lt2 < LT; ++lt2) {
      const int l0 = lt2 * 16;
      v8f s = {0.f, 0.f, 0.f, 0.f, 0.f, 0.f, 0.f, 0.f};
#pragma unroll
      for (int kc = 0; kc < 4; ++kc) {
        const int k0 = kc * 4 + 2 * hi;
        v2f a;  a.x  = sQ[k0 * STR + l0 + lh];  a.y  = sQ[(k0 + 1) * STR + l0 + lh];
        v2f bb; bb.x = sK[k0 * STR + m0 + lh];  bb.y = sK[(k0 + 1) * STR + m0 + lh];
        s = WMMA_F32(a, bb, s);
      }
      float p[8], q[8];
#pragma unroll
      for (int r = 0; r < 8; ++r) p[r] = __expf(s[r] * 0.25f - mrun);
#pragma unroll
      for (int r = 0; r < 8; ++r) q[r] = __shfl_xor(p[r], 16, 32);   // partner lane ^16
      // Repack D-layout (lane=col m, vgpr=row l) into B-layout fragments
      // (lane=col m, vgpr pair = K-chunk rows) using only the lane^16 values.
      v2f bf[4];
      bf[0].x = hi ? q[2] : p[0];  bf[0].y = hi ? q[3] : p[1];   // rows 0..3
      bf[1].x = hi ? q[6] : p[4];  bf[1].y = hi ? q[7] : p[5];   // rows 4..7
      bf[2].x = hi ? p[2] : q[0];  bf[2].y = hi ? p[3] : q[1];   // rows 8..11
      bf[3].x = hi ? p[6] : q[4];  bf[3].y = hi ? p[7] : q[5];   // rows 12..15
#pragma unroll
      for (int c = 0; c < 4; ++c) {
        const int lc = l0 + c * 4 + 2 * hi;
        v2f a; a.x = sV[lh * STR + lc]; a.y = sV[lh * STR + lc + 1];   // A = V[v, l-chunk]
        hacc = WMMA_F32(a, bf[c], hacc);
      }
    }
#pragma unroll
    for (int r = 0; r < 8; ++r)
      outp[(size_t)(r + 8 * hi) * Ln + m0 + lh] = hacc[r] * inv;
  }
}

// ---------------------------------------------------------------------------
// Kernel 2: out[b] = Wo[b][128,128] @ head[b][128,512]. Block = (b, o-tile).
// ---------------------------------------------------------------------------
__global__ __launch_bounds__(256) void outproj_kernel(
    const float* __restrict__ Wo,    // [B][128][128]
    const float* __restrict__ head,  // [B][128][512]
    float* __restrict__ out)         // [B][128][512]
{
  const int bo   = blockIdx.x;       // b*8 + o-tile
  const int b    = bo >> 3;
  const int o0   = (bo & 7) * 16;
  const int tid  = threadIdx.x;
  const int wave = tid >> 5;
  const int lane = tid & 31;
  const int lh   = lane & 15;
  const int hi   = (lane >> 4) & 1;

  const float* wo = Wo   + (size_t)b * DM * DM;
  const float* hd = head + (size_t)b * DM * Ln;
  float*       ob = out  + (size_t)b * DM * Ln;

  for (int nt = wave; nt < LT; nt += 8) {
    const int n0 = nt * 16;
    v8f acc = {0.f, 0.f, 0.f, 0.f, 0.f, 0.f, 0.f, 0.f};
    for (int d0 = 0; d0 < DM; d0 += 4) {
      const int dd = d0 + 2 * hi;
      v2f a;  a.x  = wo[(o0 + lh) * DM + dd];   a.y  = wo[(o0 + lh) * DM + dd + 1];
      v2f bb; bb.x = hd[dd * Ln + n0 + lh];     bb.y = hd[(dd + 1) * Ln + n0 + lh];
      acc = WMMA_F32(a, bb, acc);
    }
#pragma unroll
    for (int r = 0; r < 8; ++r)
      ob[(size_t)(o0 + r + 8 * hi) * Ln + n0 + lh] = acc[r];
  }
}

extern "C" void kernel_launch(void* const* d_in, const int* in_sizes, int n_in,
                              void* d_out, int out_size, void* d_ws, size_t ws_size,
                              hipStream_t stream) {
  const float* x  = (const float*)d_in[0];
  const float* Wq = (const float*)d_in[1];
  const float* Wk = (const float*)d_in[2];
  const float* Wv = (const float*)d_in[3];
  const float* Wo = (const float*)d_in[4];
  float* head = (float*)d_ws;                 // B*128*512 floats = 6.3 MB scratch
  float* outp = (float*)d_out;

  const size_t lds_bytes = (size_t)3 * DKn * STR * sizeof(float);   // 99072 B

  attn_hb_kernel<<<Hn * Bn, 256, lds_bytes, stream>>>(x, Wq, Wk, Wv, head);
  outproj_kernel<<<Bn * 8, 256, 0, stream>>>(Wo, head, outp);
}